// MultiHeadedAttention_68418829025528
// MI455X (gfx1250) — compile-verified
//
#include <hip/hip_runtime.h>

// ---------------------------------------------------------------------------
// MultiHeadedAttention (topic-gated) for MI455X / gfx1250, fp32 WMMA path.
//   context = gate*tctx + (1-gate)*ctx   (exact algebraic reduction of mix@v
//   since gate is broadcast over k and d), so we never materialize mix.
// Data movement: GLOBAL_LOAD_ASYNC_TO_LDS_B128 with double-buffered LDS tiles
// (ASYNCcnt); math: V_WMMA_F32_16X16X4_F32 throughout.
// ---------------------------------------------------------------------------

typedef __attribute__((ext_vector_type(2))) float v2f;
typedef __attribute__((ext_vector_type(4))) float v4f;
typedef __attribute__((ext_vector_type(8))) float v8f;
typedef __attribute__((ext_vector_type(4))) int   v4i;

typedef __attribute__((address_space(1))) v4i* gbl_v4i_ptr;
typedef __attribute__((address_space(3))) v4i* lds_v4i_ptr;

#define LSEQ 1024
#define HH   8
#define DHH  64
#define DD   512
#define NEGV (-1.0e18f)

#if __has_builtin(__builtin_amdgcn_global_load_async_to_lds_b128)
#define HAVE_ASYNC_LDS 1
#else
#define HAVE_ASYNC_LDS 0
#endif

static __device__ __forceinline__ v8f wmma_f32(v2f a, v2f b, v8f c) {
  // V_WMMA_F32_16X16X4_F32 : D = A(16x4) * B(4x16) + C(16x16)
  return __builtin_amdgcn_wmma_f32_16x16x4_f32(
      /*neg_a=*/false, a, /*neg_b=*/false, b,
      /*c_mod=*/(short)0, c, /*reuse_a=*/false, /*reuse_b=*/false);
}

static __device__ __forceinline__ void wait_asynccnt0() {
#if __has_builtin(__builtin_amdgcn_s_wait_asynccnt)
  __builtin_amdgcn_s_wait_asynccnt(0);
#else
  asm volatile("s_wait_asynccnt 0" ::: "memory");
#endif
}

static __device__ __forceinline__ void async_b128(const float* g, float* l) {
#if HAVE_ASYNC_LDS
  __builtin_amdgcn_global_load_async_to_lds_b128((gbl_v4i_ptr)g, (lds_v4i_ptr)l, 0, 0);
#else
  *(v4f*)l = *(const v4f*)g;
#endif
}

// ---------------------------------------------------------------------------
// Generic tiled GEMM: C[M,N] = (A[M,K] @ W[K,N] + bias[N]) * scale
// Block: 256 threads = 8 waves; block tile 64x32; each wave owns a 16x16 tile.
// GUARD=false requires M%64==0, N%32==0, K%16==0 (true for all but the
// topic_vec projection where K==300).  Fast path: async double-buffered LDS.
// ---------------------------------------------------------------------------
#define TM 64
#define TN 32
#define TK 16

template <bool GUARD>
__global__ __launch_bounds__(256) void gemm_bias_scale(
    const float* __restrict__ A, const float* __restrict__ W,
    const float* __restrict__ bias, float* __restrict__ C,
    int M, int N, int K, float scale)
{
  __shared__ float As[2][TM][TK + 4];   // +4: rows stay 16B aligned for b128
  __shared__ float Ws[2][TK][TN + 4];

  const int tid  = threadIdx.x;
  const int lane = tid & 31;
  const int wave = tid >> 5;
  const int hf   = lane >> 4;      // which half of the wave (K / row split)
  const int r    = lane & 15;

  const int mBlk = blockIdx.y * TM;
  const int nBlk = blockIdx.x * TN;
  const int wM   = (wave & 3) * 16;   // 4 wave-tiles along M
  const int wN   = (wave >> 2) * 16;  // 2 wave-tiles along N

  v8f acc = {};

  if (GUARD) {
    for (int k0 = 0; k0 < K; k0 += TK) {
      for (int i = tid; i < TM * TK; i += 256) {
        int row = i / TK, col = i % TK;
        int gr = mBlk + row, gc = k0 + col;
        As[0][row][col] = (gr < M && gc < K) ? A[(size_t)gr * K + gc] : 0.0f;
      }
      for (int i = tid; i < TK * TN; i += 256) {
        int row = i / TN, col = i % TN;
        int gr = k0 + row, gc = nBlk + col;
        Ws[0][row][col] = (gr < K && gc < N) ? W[(size_t)gr * N + gc] : 0.0f;
      }
      if (k0 + TK < K && tid < TK)
        __builtin_prefetch(&W[(size_t)(k0 + TK + tid) * N + nBlk], 0, 0);
      __syncthreads();
#pragma unroll
      for (int kk = 0; kk < TK; kk += 4) {
        v2f a, b;
        a.x = As[0][wM + r][kk + 2 * hf];
        a.y = As[0][wM + r][kk + 2 * hf + 1];
        b.x = Ws[0][kk + 2 * hf][wN + r];
        b.y = Ws[0][kk + 2 * hf + 1][wN + r];
        acc = wmma_f32(a, b, acc);
      }
      __syncthreads();
    }
  } else {
    // ---- fast path: async-to-LDS double buffering ----
    const int arow = tid >> 2, aseg = tid & 3;      // A: 256 x b128
    const int wrow = tid >> 3, wseg = tid & 7;      // W: 128 x b128
    auto stage = [&](int k0, int bufi) {
      async_b128(&A[(size_t)(mBlk + arow) * K + k0 + aseg * 4],
                 &As[bufi][arow][aseg * 4]);
      if (tid < 128)
        async_b128(&W[(size_t)(k0 + wrow) * N + nBlk + wseg * 4],
                   &Ws[bufi][wrow][wseg * 4]);
    };

    stage(0, 0);
    wait_asynccnt0();
    __syncthreads();

    int buf = 0;
    for (int k0 = 0; k0 < K; k0 += TK) {
      if (k0 + TK < K) stage(k0 + TK, buf ^ 1);   // overlap with compute
#pragma unroll
      for (int kk = 0; kk < TK; kk += 4) {
        v2f a, b;
        a.x = As[buf][wM + r][kk + 2 * hf];
        a.y = As[buf][wM + r][kk + 2 * hf + 1];
        b.x = Ws[buf][kk + 2 * hf][wN + r];
        b.y = Ws[buf][kk + 2 * hf + 1][wN + r];
        acc = wmma_f32(a, b, acc);
      }
      wait_asynccnt0();
      __syncthreads();
      buf ^= 1;
    }
  }

  const int col = nBlk + wN + r;
  if (!GUARD || col < N) {
    const float bv = bias ? bias[col] : 0.0f;
#pragma unroll
    for (int i = 0; i < 8; i++) {
      int row = mBlk + wM + i + 8 * hf;
      if (!GUARD || row < M) C[(size_t)row * N + col] = (acc[i] + bv) * scale;
    }
  }
}

// ---------------------------------------------------------------------------
// Topic scores: tsc[(b*H+h)*L + l] = dot(tv[b,l,h,:], tk[b,l,h,:]) over DH
// ---------------------------------------------------------------------------
__global__ void tscore_kernel(const float* __restrict__ TV,
                              const float* __restrict__ TKp,
                              float* __restrict__ out, int total)
{
  int idx = blockIdx.x * blockDim.x + threadIdx.x;
  if (idx >= total) return;                       // total = B*H*L
  int b   = idx / (HH * LSEQ);
  int rem = idx % (HH * LSEQ);
  int h   = rem / LSEQ;
  int l   = rem % LSEQ;
  const float* a = TV  + ((size_t)(b * LSEQ + l)) * DD + h * DHH;
  const float* c = TKp + ((size_t)(b * LSEQ + l)) * DD + h * DHH;
  float s = 0.0f;
#pragma unroll 8
  for (int d = 0; d < DHH; d++) s += a[d] * c[d];
  out[idx] = s;
}

// ---------------------------------------------------------------------------
// Flash-style attention with dual online softmax (attn + topic-attn).
// Block = 256 threads (8 waves), one (b,h), 128 q rows; each wave owns 16 q.
// K/V tiles: double-buffered GLOBAL_LOAD_ASYNC_TO_LDS_B128 (next chunk's
// fetch overlaps the current chunk's 48 WMMAs per wave).
// ---------------------------------------------------------------------------
__global__ __launch_bounds__(256) void attn_kernel(
    const float* __restrict__ Qp, const float* __restrict__ Kp,
    const float* __restrict__ Vp, const float* __restrict__ tsc,
    const unsigned char* __restrict__ mask,
    float* __restrict__ ctxO, float* __restrict__ tctxO)
{
  __shared__ float Kc[2][16][DHH + 4];  // +4 keeps rows 16B-aligned for b128
  __shared__ float Vc[2][16][DHH + 4];
  __shared__ float Pl[8][16][17];

  const int tid  = threadIdx.x;
  const int lane = tid & 31;
  const int wave = tid >> 5;
  const int hf   = lane >> 4;
  const int r    = lane & 15;

  const int blocksPerBH = LSEQ / 128;
  const int bh = blockIdx.x / blocksPerBH;
  const int qc = blockIdx.x % blocksPerBH;
  const int b  = bh / HH, h = bh % HH;
  const int qBase = qc * 128 + wave * 16;

  // resident Q fragments: 16 chunks of K=4 over DH=64 (already * 1/sqrt(dh))
  v2f qf[16];
  {
    const float* qrow = Qp + ((size_t)(b * LSEQ + qBase + r)) * DD + h * DHH;
#pragma unroll
    for (int dc = 0; dc < 16; dc++) {
      qf[dc].x = qrow[dc * 4 + 2 * hf];
      qf[dc].y = qrow[dc * 4 + 2 * hf + 1];
    }
  }

  v8f acc[4] = {{0}}, tacc[4] = {{0}};
  float m[8], lsum[8], tm[8], tl[8];
#pragma unroll
  for (int i = 0; i < 8; i++) { m[i] = -3.0e38f; tm[i] = -3.0e38f; lsum[i] = 0.0f; tl[i] = 0.0f; }

  const int krow = tid >> 4, kseg = tid & 15;   // 256 x b128 per K (and V) tile
  auto stageKV = [&](int kb, int bufi) {
    size_t g = ((size_t)(b * LSEQ + kb + krow)) * DD + h * DHH + kseg * 4;
    async_b128(Kp + g, &Kc[bufi][krow][kseg * 4]);
    async_b128(Vp + g, &Vc[bufi][krow][kseg * 4]);
  };

  stageKV(0, 0);
  wait_asynccnt0();
  __syncthreads();

  int buf = 0;
  for (int kb = 0; kb < LSEQ; kb += 16) {
    if (kb + 16 < LSEQ) stageKV(kb + 16, buf ^ 1);   // overlap with compute

    // ---- S = Q K^T (16x16) ----
    v8f s = {};
#pragma unroll
    for (int dc = 0; dc < 16; dc++) {
      v2f bf;
      bf.x = Kc[buf][r][dc * 4 + 2 * hf];
      bf.y = Kc[buf][r][dc * 4 + 2 * hf + 1];
      s = wmma_f32(qf[dc], bf, s);
    }

    // ---- mask + topic score tile (per-lane coalesced loads) ----
    const float tsv = tsc[((size_t)(b * HH + h)) * LSEQ + kb + r];
    v8f ts;
#pragma unroll
    for (int i = 0; i < 8; i++) {
      int qrow = qBase + i + 8 * hf;
      bool mk = mask[((size_t)b * LSEQ + qrow) * LSEQ + kb + r] != 0;
      ts[i] = mk ? NEGV : tsv;
      if (mk) s[i] = NEGV;
    }

    // ---- dual online softmax; P values left in s / ts ----
#pragma unroll
    for (int i = 0; i < 8; i++) {
      float rm = s[i];
#pragma unroll
      for (int o = 1; o < 16; o <<= 1) rm = fmaxf(rm, __shfl_xor(rm, o, 16));
      float nm = fmaxf(m[i], rm);
      float fs = __expf(m[i] - nm);
      float p  = __expf(s[i] - nm);
      float rs = p;
#pragma unroll
      for (int o = 1; o < 16; o <<= 1) rs += __shfl_xor(rs, o, 16);
      lsum[i] = lsum[i] * fs + rs;
      m[i] = nm; s[i] = p;
#pragma unroll
      for (int n = 0; n < 4; n++) acc[n][i] *= fs;

      float trm = ts[i];
#pragma unroll
      for (int o = 1; o < 16; o <<= 1) trm = fmaxf(trm, __shfl_xor(trm, o, 16));
      float tnm = fmaxf(tm[i], trm);
      float tfs = __expf(tm[i] - tnm);
      float tp  = __expf(ts[i] - tnm);
      float trs = tp;
#pragma unroll
      for (int o = 1; o < 16; o <<= 1) trs += __shfl_xor(trs, o, 16);
      tl[i] = tl[i] * tfs + trs;
      tm[i] = tnm; ts[i] = tp;
#pragma unroll
      for (int n = 0; n < 4; n++) tacc[n][i] *= tfs;
    }

    // ---- ctx += P @ V ----  (transpose P through wave-private LDS slab)
#pragma unroll
    for (int i = 0; i < 8; i++) Pl[wave][i + 8 * hf][r] = s[i];
    __builtin_amdgcn_wave_barrier();
    {
      v2f af[4];
#pragma unroll
      for (int st = 0; st < 4; st++) {
        af[st].x = Pl[wave][r][st * 4 + 2 * hf];
        af[st].y = Pl[wave][r][st * 4 + 2 * hf + 1];
      }
#pragma unroll
      for (int n = 0; n < 4; n++)
#pragma unroll
        for (int st = 0; st < 4; st++) {
          v2f bf;
          bf.x = Vc[buf][st * 4 + 2 * hf][n * 16 + r];
          bf.y = Vc[buf][st * 4 + 2 * hf + 1][n * 16 + r];
          acc[n] = wmma_f32(af[st], bf, acc[n]);
        }
    }
    __builtin_amdgcn_wave_barrier();

    // ---- tctx += tP @ V ----
#pragma unroll
    for (int i = 0; i < 8; i++) Pl[wave][i + 8 * hf][r] = ts[i];
    __builtin_amdgcn_wave_barrier();
    {
      v2f af[4];
#pragma unroll
      for (int st = 0; st < 4; st++) {
        af[st].x = Pl[wave][r][st * 4 + 2 * hf];
        af[st].y = Pl[wave][r][st * 4 + 2 * hf + 1];
      }
#pragma unroll
      for (int n = 0; n < 4; n++)
#pragma unroll
        for (int st = 0; st < 4; st++) {
          v2f bf;
          bf.x = Vc[buf][st * 4 + 2 * hf][n * 16 + r];
          bf.y = Vc[buf][st * 4 + 2 * hf + 1][n * 16 + r];
          tacc[n] = wmma_f32(af[st], bf, tacc[n]);
        }
    }

    wait_asynccnt0();       // next chunk's tiles landed
    __syncthreads();        // everyone done reading buf
    buf ^= 1;
  }

  // finalize: divide by softmax sums, write (B,L,H*DH) "unshaped" layout
#pragma unroll
  for (int n = 0; n < 4; n++)
#pragma unroll
    for (int i = 0; i < 8; i++) {
      int qrow = qBase + i + 8 * hf;
      size_t g = ((size_t)(b * LSEQ + qrow)) * DD + h * DHH + n * 16 + r;
      ctxO[g]  = acc[n][i] / lsum[i];
      tctxO[g] = tacc[n][i] / tl[i];
    }
}

// ---------------------------------------------------------------------------
// gate[b,l,h] = sigmoid( [q|ctx|tctx](b,l,:) @ Wtw[:,h] + btw[h] )
// ---------------------------------------------------------------------------
__global__ void gate_kernel(const float* __restrict__ Qp,
                            const float* __restrict__ ctx,
                            const float* __restrict__ tctx,
                            const float* __restrict__ Wtw,
                            const float* __restrict__ btw,
                            float* __restrict__ gate, int BL)
{
  int idx = blockIdx.x * blockDim.x + threadIdx.x;
  if (idx >= BL * HH) return;
  int row = idx / HH, h = idx % HH;
  float s = btw[h];
  const float* q = Qp   + (size_t)row * DD;
  const float* c = ctx  + (size_t)row * DD;
  const float* t = tctx + (size_t)row * DD;
  for (int f = 0; f < DD; f++) s += q[f] * Wtw[(size_t)f * HH + h];
  for (int f = 0; f < DD; f++) s += c[f] * Wtw[(size_t)(DD + f) * HH + h];
  for (int f = 0; f < DD; f++) s += t[f] * Wtw[(size_t)(2 * DD + f) * HH + h];
  gate[idx] = 1.0f / (1.0f + __expf(-s));
}

// ---------------------------------------------------------------------------
// mixed context = gate*tctx + (1-gate)*ctx   (elementwise, head-indexed gate)
// ---------------------------------------------------------------------------
__global__ void mix_kernel(const float* __restrict__ ctx,
                           const float* __restrict__ tctx,
                           const float* __restrict__ gate,
                           float* __restrict__ out, int total)
{
  int idx = blockIdx.x * blockDim.x + threadIdx.x;
  if (idx >= total) return;
  int row = idx / DD, col = idx % DD;
  int h = col / DHH;
  float g = gate[(size_t)row * HH + h];
  out[idx] = g * tctx[idx] + (1.0f - g) * ctx[idx];
}

// ---------------------------------------------------------------------------
extern "C" void kernel_launch(void* const* d_in, const int* in_sizes, int n_in,
                              void* d_out, int out_size, void* d_ws, size_t ws_size,
                              hipStream_t stream)
{
  const float* key   = (const float*)d_in[0];
  const float* value = (const float*)d_in[1];
  const float* query = (const float*)d_in[2];
  const float* topic = (const float*)d_in[3];
  const unsigned char* mask = (const unsigned char*)d_in[4];
  const float* Wk  = (const float*)d_in[5];
  const float* bk  = (const float*)d_in[6];
  const float* Wv  = (const float*)d_in[7];
  const float* bv  = (const float*)d_in[8];
  const float* Wq  = (const float*)d_in[9];
  const float* bq  = (const float*)d_in[10];
  const float* Wtk = (const float*)d_in[11];
  const float* btk = (const float*)d_in[12];
  const float* Wtv = (const float*)d_in[13];
  const float* btv = (const float*)d_in[14];
  const float* Wtw = (const float*)d_in[15];
  const float* btw = (const float*)d_in[16];
  const float* Wo  = (const float*)d_in[17];
  const float* bo  = (const float*)d_in[18];

  const int Bn = in_sizes[0] / (LSEQ * DD);    // 4
  const int M  = Bn * LSEQ;                    // 4096
  const int TDdim = in_sizes[3] / (Bn * LSEQ); // 300
  const float INV = 0.125f;                    // 1/sqrt(64)

  // workspace layout (floats)
  float* ws = (float*)d_ws;
  const size_t P = (size_t)M * DD;
  float* Kp    = ws;
  float* Vp    = Kp + P;
  float* Qp    = Vp + P;
  float* TKp   = Qp + P;
  float* TVp   = TKp + P;
  float* tsc   = TVp + P;                 // B*H*L
  float* gateb = tsc + (size_t)Bn * HH * LSEQ;
  float* ctx   = gateb + (size_t)M * HH;
  float* tctx  = ctx + P;
  float* mixb  = tctx + P;

  dim3 gblk(256);
  dim3 ggrid(DD / TN, M / TM);

  gemm_bias_scale<false><<<ggrid, gblk, 0, stream>>>(key,   Wk,  bk,  Kp,  M, DD, DD,    1.0f);
  gemm_bias_scale<false><<<ggrid, gblk, 0, stream>>>(value, Wv,  bv,  Vp,  M, DD, DD,    1.0f);
  gemm_bias_scale<false><<<ggrid, gblk, 0, stream>>>(query, Wq,  bq,  Qp,  M, DD, DD,    INV);
  gemm_bias_scale<false><<<ggrid, gblk, 0, stream>>>(key,   Wtk, btk, TKp, M, DD, DD,    1.0f);
  gemm_bias_scale<true ><<<ggrid, gblk, 0, stream>>>(topic, Wtv, btv, TVp, M, DD, TDdim, INV);

  int tTotal = Bn * HH * LSEQ;
  tscore_kernel<<<(tTotal + 255) / 256, 256, 0, stream>>>(TVp, TKp, tsc, tTotal);

  attn_kernel<<<Bn * HH * (LSEQ / 128), 256, 0, stream>>>(Qp, Kp, Vp, tsc, mask, ctx, tctx);

  gate_kernel<<<(M * HH + 255) / 256, 256, 0, stream>>>(Qp, ctx, tctx, Wtw, btw, gateb, M);

  mix_kernel<<<((int)P + 255) / 256, 256, 0, stream>>>(ctx, tctx, gateb, mixb, (int)P);

  gemm_bias_scale<false><<<ggrid, gblk, 0, stream>>>(mixb, Wo, bo, (float*)d_out, M, DD, DD, 1.0f);
}